// StatDenoiser_14791867368130
// MI455X (gfx1250) — compile-verified
//
#include <hip/hip_runtime.h>

// ---------------- problem constants (match reference setup_inputs) ----------
constexpr int H = 256, W = 256, T = 32;
constexpr int CH = H * W * T;             // per-channel element count
constexpr int SR = 3, TR = 1;             // spatial / temporal radius
constexpr int BH = 8, BW = 8, BT = 8;     // output tile per workgroup
constexpr int PH = BH + 2 * SR;           // 14
constexpr int PW = BW + 2 * SR;           // 14
constexpr int PT = BT + 2 * TR;           // 10
constexpr int NPOS = PH * PW * PT;        // 1960 padded positions
constexpr int SLOTS = 20;                 // g0..8 | q | e0..2 | g2v0..2 | n0..2 | pad
constexpr int NT = 256;                   // 8 waves (wave32)
constexpr float GAMMA_W = 2.9146f;
constexpr float G2 = GAMMA_W * GAMMA_W;   // gamma^2 (t-test folded to squared form)
constexpr float EPS = 1e-10f;

#define AS1 __attribute__((address_space(1)))
#define AS3 __attribute__((address_space(3)))

#if defined(__has_builtin)
#if __has_builtin(__builtin_amdgcn_global_load_async_to_lds_b32)
#define HAS_ASYNC_LDS 1
#endif
#if __has_builtin(__builtin_amdgcn_s_wait_asynccnt)
#define HAS_WAIT_ASYNC 1
#endif
#endif

__device__ __forceinline__ void async_cp(const float* g, float* l) {
#if defined(HAS_ASYNC_LDS)
  // CDNA5 async copy: bypasses VGPRs, lands directly in LDS, tracked by ASYNCcnt
  __builtin_amdgcn_global_load_async_to_lds_b32((AS1 int*)g, (AS3 int*)l,
                                                /*imm offset*/ 0, /*cpol*/ 0);
#else
  *l = *g;  // fallback: register-staged copy
#endif
}

__device__ __forceinline__ void wait_async() {
#if defined(HAS_ASYNC_LDS)
#if defined(HAS_WAIT_ASYNC)
  __builtin_amdgcn_s_wait_asynccnt(0);
#else
  asm volatile("s_wait_asynccnt 0" ::: "memory");
#endif
#endif
}

__global__ __launch_bounds__(NT) void statden_kernel(
    const float* __restrict__ noisy, const float* __restrict__ guidance,
    const float* __restrict__ estim, const float* __restrict__ evar,
    const float* __restrict__ sigma_inv, float* __restrict__ out) {
  extern __shared__ float sm[];  // NPOS * SLOTS floats = 156.8 KB (2 WG / WGP)
  const int tid = threadIdx.x;
  const int t0 = blockIdx.x * BT;
  const int w0 = blockIdx.y * BW;
  const int h0 = blockIdx.z * BH;

  float sig[9];
#pragma unroll
  for (int c = 0; c < 9; ++c) sig[c] = sigma_inv[c];

  // ---- stage padded 18-channel tile into LDS (zero OOB halo = reference pad) ----
  for (int i = tid; i < NPOS; i += NT) {
    const int ph = i / (PW * PT);
    const int r = i - ph * (PW * PT);
    const int pw = r / PT;
    const int pt = r - pw * PT;
    const int gh = h0 + ph - SR, gw = w0 + pw - SR, gt = t0 + pt - TR;
    float* lp = sm + i * SLOTS;
    if ((unsigned)gh < (unsigned)H && (unsigned)gw < (unsigned)W &&
        (unsigned)gt < (unsigned)T) {
      const size_t go = ((size_t)gh * W + gw) * T + gt;
#pragma unroll
      for (int c = 0; c < 9; ++c) async_cp(guidance + (size_t)c * CH + go, lp + c);
#pragma unroll
      for (int c = 0; c < 3; ++c) async_cp(estim + (size_t)c * CH + go, lp + 10 + c);
#pragma unroll
      for (int c = 0; c < 3; ++c) async_cp(evar + (size_t)c * CH + go, lp + 13 + c);
#pragma unroll
      for (int c = 0; c < 3; ++c) async_cp(noisy + (size_t)c * CH + go, lp + 16 + c);
    } else {
      const float4 z = make_float4(0.f, 0.f, 0.f, 0.f);
      float4* z4 = reinterpret_cast<float4*>(lp);
#pragma unroll
      for (int k = 0; k < 5; ++k) z4[k] = z;
    }
  }
  wait_async();      // each wave drains its own ASYNCcnt
  __syncthreads();   // then workgroup barrier makes LDS visible

  // ---- per-position precompute: q = sum_c sig_c*g_c^2 ; v *= gamma^2 (in place) ----
  for (int i = tid; i < NPOS; i += NT) {
    float* lp = sm + i * SLOTS;
    float q = 0.f;
#pragma unroll
    for (int c = 0; c < 9; ++c) { const float g = lp[c]; q += sig[c] * g * g; }
    lp[9] = q;
#pragma unroll
    for (int c = 0; c < 3; ++c) lp[13 + c] *= G2;
  }
  __syncthreads();

  // ---- stencil: lanes t-fastest (bank-friendly b128), 2 pixels per thread ----
  const int tl0 = tid & 3;
  const int wl = (tid >> 2) & 7;
  const int hl = tid >> 5;

  for (int half = 0; half < 2; ++half) {
    const int tl = tl0 + 4 * half;
    const float* cp = sm + (((hl + SR) * PW + (wl + SR)) * PT + (tl + TR)) * SLOTS;
    float sg[9];
#pragma unroll
    for (int c = 0; c < 9; ++c) sg[c] = sig[c] * cp[c];
    const float ci = -cp[9];
    const float e0 = cp[10], e1 = cp[11], e2 = cp[12];
    const float r0 = cp[13] + G2 * EPS;
    const float r1 = cp[14] + G2 * EPS;
    const float r2 = cp[15] + G2 * EPS;

    float num0 = 0.f, num1 = 0.f, num2 = 0.f, den = 0.f;
#pragma unroll 1
    for (int dh = 0; dh < 2 * SR + 1; ++dh) {
      const int rowb = (hl + dh) * PW;
#pragma unroll 1
      for (int dw = 0; dw < 2 * SR + 1; ++dw) {
        const int posb = (rowb + (wl + dw)) * PT + tl;
#pragma unroll
        for (int dt = 0; dt < 2 * TR + 1; ++dt) {
          const float4* p4 = reinterpret_cast<const float4*>(sm + (posb + dt) * SLOTS);
          const float4 a = p4[0];    // g0..g3
          const float4 b = p4[1];    // g4..g7
          const float4 cq = p4[2];   // g8, q, e0, e1
          const float4 dv = p4[3];   // e2, g2v0, g2v1, g2v2
          const float4 nn = p4[4];   // n0, n1, n2, (pad)
          float dot = sg[0] * a.x;
          dot += sg[1] * a.y; dot += sg[2] * a.z; dot += sg[3] * a.w;
          dot += sg[4] * b.x; dot += sg[5] * b.y; dot += sg[6] * b.z;
          dot += sg[7] * b.w; dot += sg[8] * cq.x;
          // -(q_j + q_i - 2*dot) == -sum_c sig_c*(g_j - g_i)^2
          const float arg = 2.f * dot + ci - cq.y;
          float wgt = __expf(arg);
          const float d0 = cq.z - e0, d1 = cq.w - e1, d2 = dv.x - e2;
          // t-test, squared form: de^2 <= gamma^2*(v_j + v_i + eps)
          const bool mem = (d0 * d0 <= dv.y + r0) & (d1 * d1 <= dv.z + r1) &
                           (d2 * d2 <= dv.w + r2);
          wgt = mem ? wgt : 0.f;
          num0 += wgt * nn.x; num1 += wgt * nn.y; num2 += wgt * nn.z;
          den += wgt;
        }
      }
    }
    const float inv = 1.f / (den + EPS);
    const size_t o = ((size_t)(h0 + hl) * W + (w0 + wl)) * T + (t0 + tl);
    out[o] = num0 * inv;
    out[(size_t)CH + o] = num1 * inv;
    out[2 * (size_t)CH + o] = num2 * inv;
  }
}

extern "C" void kernel_launch(void* const* d_in, const int* in_sizes, int n_in,
                              void* d_out, int out_size, void* d_ws, size_t ws_size,
                              hipStream_t stream) {
  const float* noisy = (const float*)d_in[0];
  const float* guidance = (const float*)d_in[1];
  const float* estim = (const float*)d_in[2];
  const float* evar = (const float*)d_in[3];
  const float* sigma_inv = (const float*)d_in[4];
  float* out = (float*)d_out;

  const size_t shmem = (size_t)NPOS * SLOTS * sizeof(float);  // 156,800 B
  static_assert((size_t)NPOS * SLOTS * sizeof(float) <= 160u * 1024u,
                "keep two workgroups resident per WGP (320 KB LDS pool)");
  (void)hipFuncSetAttribute((const void*)statden_kernel,
                            hipFuncAttributeMaxDynamicSharedMemorySize, (int)shmem);
  dim3 grid(T / BT, W / BW, H / BH);  // (4, 32, 32) tiles
  statden_kernel<<<grid, NT, shmem, stream>>>(noisy, guidance, estim, evar,
                                              sigma_inv, out);
}